// MetaSR_82506321756893
// MI455X (gfx1250) — compile-verified
//
#include <hip/hip_runtime.h>

typedef __attribute__((ext_vector_type(2))) float v2f;
typedef __attribute__((ext_vector_type(8))) float v8f;

#define HID   256
#define OUTW  1728          // 64*9*3
#define LDSTR 68            // padded channel stride (bank-conflict-free)

// ---------------------------------------------------------------------------
// Kernel 1: 16-phase MLP -> phase weight matrices, stored WMMA-fragment-ready.
// Bws[(kperm>>1)*96 + 2*n + (kperm&1)] = pred_w[p][c9][rgb]
//   kperm = (ki*3+kj)*64 + c,  n = p*3 + rgb,  c9 = c*9 + ki*3 + kj
// ---------------------------------------------------------------------------
__global__ void metasr_phase_mlp(const float* __restrict__ w1,
                                 const float* __restrict__ b1,
                                 const float* __restrict__ w2,
                                 const float* __restrict__ b2,
                                 float* __restrict__ Bws) {
  __shared__ float hbuf[HID];
  const int p = blockIdx.x;          // phase 0..15
  const int j = threadIdx.x;         // 0..255
  const float rel0 = (float)(p >> 2) * 0.25f;
  const float rel1 = (float)(p & 3) * 0.25f;
  float hv = b1[j] + rel0 * w1[j] + rel1 * w1[HID + j] + 0.25f * w1[2 * HID + j];
  hbuf[j] = hv > 0.0f ? hv : 0.0f;
  __syncthreads();
  for (int s = 0; s < 7; ++s) {
    int o = j + (s << 8);
    if (o < OUTW) {
      float acc = b2[o];
      for (int k = 0; k < HID; ++k)
        acc = fmaf(hbuf[k], w2[k * OUTW + o], acc);
      int c9  = o / 3;
      int rgb = o - c9 * 3;
      int c   = c9 / 9;
      int r   = c9 - c * 9;          // ki*3+kj
      int kperm = r * 64 + c;
      int n = p * 3 + rgb;
      Bws[(kperm >> 1) * 96 + 2 * n + (kperm & 1)] = acc;
    }
  }
}

// ---------------------------------------------------------------------------
// Kernel 2: sub-pixel 3x3 conv as WMMA f32 GEMM.
// Block = one LR row iy (64 LR pixels), 4 waves x 16 ix each, N = 48.
// ---------------------------------------------------------------------------
__global__ void metasr_conv(const float* __restrict__ feat,
                            const float* __restrict__ Bws,
                            float* __restrict__ out) {
  __shared__ float fl[3 * 66 * LDSTR];       // [ki][col(=x+1)][c] padded
  const int iy  = blockIdx.x;
  const int tid = threadIdx.x;

  // Fill LDS slab: fl[(ki*66+col)*LDSTR + c] = feat[c][iy+ki-1][col-1] (0-pad)
  for (int e = tid; e < 3 * 66 * 64; e += 128) {   // 12672 = 99*128, no tail
    int c   = e / 198;
    int rem = e - c * 198;
    int ki  = rem / 66;
    int col = rem - ki * 66;
    int y = iy + ki - 1;
    int x = col - 1;
    float v = 0.0f;
    if ((unsigned)y < 64u && (unsigned)x < 64u)
      v = feat[c * 4096 + y * 64 + x];
    fl[(ki * 66 + col) * LDSTR + c] = v;
  }
  __syncthreads();

  const int wv     = tid >> 5;       // wave 0..3
  const int lane   = tid & 31;
  const int mlane  = lane & 15;      // M (ix) / N column within tile
  const int hi     = lane >> 4;      // K-pair selector
  const int ixbase = wv << 4;

  v8f acc[3];
  acc[0] = (v8f){0.f,0.f,0.f,0.f,0.f,0.f,0.f,0.f};
  acc[1] = acc[0];
  acc[2] = acc[0];

  // per-lane B fragment base: row t holds 96 floats; t = kbase/2 + hi
  const float* bbase = Bws + hi * 96 + 2 * mlane;

  for (int r = 0; r < 9; ++r) {                // (ki,kj)
    int ki = r / 3;
    int kj = r - ki * 3;
    const float* abase = fl + ((ki * 66) + ixbase + mlane + kj) * LDSTR + 2 * hi;
    const float* bptr  = bbase + (r * 32) * 96;
#pragma unroll
    for (int cb = 0; cb < 16; ++cb) {          // cbase = 4*cb, K-step of 4
      v2f a  = *(const v2f*)(abase + 4 * cb);
      v2f b0 = *(const v2f*)(bptr + 0);
      v2f b1 = *(const v2f*)(bptr + 32);
      v2f b2 = *(const v2f*)(bptr + 64);
      acc[0] = __builtin_amdgcn_wmma_f32_16x16x4_f32(
          false, a, false, b0, (short)0, acc[0], false, false);
      acc[1] = __builtin_amdgcn_wmma_f32_16x16x4_f32(
          false, a, false, b1, (short)0, acc[1], false, false);
      acc[2] = __builtin_amdgcn_wmma_f32_16x16x4_f32(
          false, a, false, b2, (short)0, acc[2], false, false);
      bptr += 2 * 96;                          // t advances by 2 per step
    }
  }

  // Scatter C[16][48] to HR output [3][256][256] (pixel shuffle).
  const int qybase = iy << 2;
#pragma unroll
  for (int nt = 0; nt < 3; ++nt) {
    int n   = (nt << 4) + mlane;
    int p   = n / 3;
    int rgb = n - p * 3;
    int dy  = p >> 2;
    int dx  = p & 3;
    float* obase = out + rgb * 65536 + (qybase + dy) * 256 + dx;
#pragma unroll
    for (int v = 0; v < 8; ++v) {
      int m  = v + (hi << 3);                  // C/D layout: lanes16-31 -> M+8
      int qx = (ixbase + m) << 2;
      obase[qx] = acc[nt][v];
    }
  }
}

extern "C" void kernel_launch(void* const* d_in, const int* in_sizes, int n_in,
                              void* d_out, int out_size, void* d_ws, size_t ws_size,
                              hipStream_t stream) {
  const float* feat = (const float*)d_in[0];
  const float* w1   = (const float*)d_in[1];
  const float* b1   = (const float*)d_in[2];
  const float* w2   = (const float*)d_in[3];
  const float* b2   = (const float*)d_in[4];
  float* out = (float*)d_out;
  float* Bws = (float*)d_ws;   // 288*96 floats = 110,592 bytes

  metasr_phase_mlp<<<16, 256, 0, stream>>>(w1, b1, w2, b2, Bws);
  metasr_conv<<<64, 128, 0, stream>>>(feat, Bws, out);
}